// FilterTransform_81398220194368
// MI455X (gfx1250) — compile-verified
//
#include <hip/hip_runtime.h>
#include <cstdint>

typedef uint32_t u32;
typedef __attribute__((ext_vector_type(4))) u32 u32x4;
typedef __attribute__((ext_vector_type(8))) u32 u32x8;

#define B_ROWS 64
#define T_LEN 480000
#define L_CHUNK 600
#define CHUNKS_PER_ROW (T_LEN / L_CHUNK)        // 800
#define TOTAL_CHUNKS (B_ROWS * CHUNKS_PER_ROW)  // 51200
#define T_TILE 25
#define NTILES (L_CHUNK / T_TILE)               // 24
#define NCHUNK_BLK 128
#define THREADS NCHUNK_BLK
#define NBLOCKS (TOTAL_CHUNKS / NCHUNK_BLK)     // 400
#define WARM 192                                 // pole radius 0.824 -> r^192 ~ 7e-17

static_assert(CHUNKS_PER_ROW * L_CHUNK == T_LEN, "chunking must tile the row");
static_assert(NTILES * T_TILE == L_CHUNK, "tiles must tile the chunk");
static_assert(NBLOCKS * NCHUNK_BLK == TOTAL_CHUNKS, "blocks must tile the chunks");

// ---- TDM descriptor builders (CDNA5 ISA ch.8, 2D tensor: groups 0 and 1 only) ----
// Group 0 (128b): [1:0]=count=1, [63:32]=lds_addr, [120:64]=global_addr, [127:126]=type=2
__device__ __forceinline__ u32x4 tdm_group0(u32 lds_byte_off, uint64_t gaddr) {
  u32x4 g;
  g[0] = 1u;                                        // count=1, user mode, no gather
  g[1] = lds_byte_off;                              // LDS byte address (wg-relative)
  g[2] = (u32)(gaddr & 0xFFFFFFFFull);              // global_addr[31:0]
  g[3] = (u32)((gaddr >> 32) & 0x01FFFFFFull)       // global_addr[56:32]
         | (2u << 30);                              // type = 2 ("image")
  return g;
}
// Group 1 (256b): wg_mask=0, data_size=4B, tile = T_TILE x NCHUNK_BLK,
// tensor dims huge (never clipped), dim0 stride = L_CHUNK elements.
__device__ __forceinline__ u32x8 tdm_group1() {
  u32x8 g;
  g[0] = (2u << 16);                       // data_size=2 (4 bytes); no pad/iterate/multicast
  g[1] = 0u;                               // atomic_barrier_addr=0; tensor_dim0[15:0]=0
  g[2] = 0x4000u;                          // tensor_dim0[31:16] (=0x40000000); tdim1 lo16=0
  g[3] = 0x4000u | ((u32)T_TILE << 16);    // tensor_dim1[31:16]; tile_dim0 = 25
  g[4] = (u32)NCHUNK_BLK;                  // tile_dim1 = 128; tile_dim2 = 0
  g[5] = (u32)L_CHUNK;                     // tensor_dim0_stride lo32 = 600
  g[6] = 0u;                               // stride0 hi16 = 0; stride1 lo16 = 0
  g[7] = 0u;                               // stride1 hi32 = 0
  return g;
}

__device__ __forceinline__ void tdm_load(u32x4 g0, u32x8 g1) {
  asm volatile("tensor_load_to_lds %0, %1" :: "s"(g0), "s"(g1) : "memory");
}
__device__ __forceinline__ void tdm_store(u32x4 g0, u32x8 g1) {
  asm volatile("tensor_store_from_lds %0, %1" :: "s"(g0), "s"(g1) : "memory");
}

__global__ __launch_bounds__(THREADS)
void biquad_eq_tdm_kernel(const float* __restrict__ x, float* __restrict__ y) {
  // Load: triple-buffered, Store: double-buffered. Dense rows of 25 dwords:
  // bank(j*25+t) = (25j+t) mod 64 is conflict-free across a wave (25 odd).
  __shared__ float ldsIn [3][NCHUNK_BLK * T_TILE];
  __shared__ float ldsOut[2][NCHUNK_BLK * T_TILE];

  // ---- equalizer_biquad coefficients (compile-time, double then cast, like numpy) ----
  constexpr double kSinW0 = 0.38268343236508978;   // sin(2*pi*1000/16000) = sin(pi/8)
  constexpr double kCosW0 = 0.92387953251128674;   // cos(pi/8)
  constexpr double kA     = 1.4125375446227544;    // 10^(6/40)
  constexpr double kAlpha = kSinW0 / (2.0 * 0.707);
  constexpr double a0d = 1.0 + kAlpha / kA;
  const float B0 = (float)((1.0 + kAlpha * kA) / a0d);
  const float B1 = (float)((-2.0 * kCosW0) / a0d);
  const float B2 = (float)((1.0 - kAlpha * kA) / a0d);
  const float A1 = (float)((-2.0 * kCosW0) / a0d);
  const float A2 = (float)((1.0 - kAlpha / kA) / a0d);

  const int j = threadIdx.x;
  const long long gchunk0 = (long long)blockIdx.x * NCHUNK_BLK;
  const long long g = gchunk0 + j;                 // this lane's chunk id
  const long long start = g * (long long)L_CHUNK;  // flat sample index of chunk start

  // Wave-0-of-block leads all TDM traffic (scalar-uniform condition -> s_cbranch).
  const bool lead = (__builtin_amdgcn_readfirstlane((int)threadIdx.x) < 32);

  const u32 inOff[3]  = { (u32)(uintptr_t)&ldsIn[0][0],
                          (u32)(uintptr_t)&ldsIn[1][0],
                          (u32)(uintptr_t)&ldsIn[2][0] };
  const u32 outOff[2] = { (u32)(uintptr_t)&ldsOut[0][0],
                          (u32)(uintptr_t)&ldsOut[1][0] };

  const uint64_t xbase = (uint64_t)(uintptr_t)(x + gchunk0 * (long long)L_CHUNK);
  const uint64_t ybase = (uint64_t)(uintptr_t)(y + gchunk0 * (long long)L_CHUNK);
  const u32x8 g1 = tdm_group1();

  // Prologue: prime 3 tile loads (in-flight pipeline).
  if (lead) {
    tdm_load(tdm_group0(inOff[0], xbase + 0ull * (T_TILE * 4)), g1);
    tdm_load(tdm_group0(inOff[1], xbase + 1ull * (T_TILE * 4)), g1);
    tdm_load(tdm_group0(inOff[2], xbase + 2ull * (T_TILE * 4)), g1);
  }

  // ---- Warm-up: run recurrence WARM samples before the chunk with zero y-state.
  // Exact x-history is read from memory; rows start with true zero state.
  float x1 = 0.f, x2 = 0.f, y1 = 0.f, y2 = 0.f;
  if ((g % (long long)CHUNKS_PER_ROW) != 0) {
    const float* xp = x + (start - (long long)WARM);
    x2 = xp[-2];
    x1 = xp[-1];
#pragma unroll 4
    for (int n = 0; n < WARM; ++n) {
      float xt = xp[n];
      float yv = B0 * xt + B1 * x1 + B2 * x2 - A1 * y1 - A2 * y2;
      x2 = x1; x1 = xt; y2 = y1; y1 = yv;
    }
  }

  // ---- Main pipeline over 24 tiles of 25 samples.
  for (int k = 0; k < NTILES; ++k) {
    const int kb3 = k % 3;
    const int kb2 = k & 1;
    if (lead) {
      // In-order TENSORcnt: <=2 guarantees load k done and store k-2 drained
      // for every k except the last, which also needs <=1 (see issue schedule).
      __builtin_amdgcn_s_wait_tensorcnt(2);
      if (k == NTILES - 1) __builtin_amdgcn_s_wait_tensorcnt(1);
    }
    __syncthreads();  // tile k ready in ldsIn[kb3]; ldsOut[kb2] free

    const int base = j * T_TILE;
#pragma unroll
    for (int t = 0; t < T_TILE; ++t) {
      float xt = ldsIn[kb3][base + t];
      float yv = B0 * xt + B1 * x1 + B2 * x2 - A1 * y1 - A2 * y2;
      ldsOut[kb2][base + t] = yv;
      x2 = x1; x1 = xt; y2 = y1; y1 = yv;
    }
    __syncthreads();  // all lanes finished ldsOut[kb2] / done reading ldsIn[kb3]

    if (lead) {
      tdm_store(tdm_group0(outOff[kb2], ybase + (uint64_t)k * (T_TILE * 4)), g1);
      if (k + 3 < NTILES)
        tdm_load(tdm_group0(inOff[kb3], xbase + (uint64_t)(k + 3) * (T_TILE * 4)), g1);
    }
  }
  // S_ENDPGM waits for all counters (incl. TENSORcnt) -> final stores complete.
}

extern "C" void kernel_launch(void* const* d_in, const int* in_sizes, int n_in,
                              void* d_out, int out_size, void* d_ws, size_t ws_size,
                              hipStream_t stream) {
  const float* x = (const float*)d_in[0];
  float* y = (float*)d_out;
  (void)in_sizes; (void)n_in; (void)out_size; (void)d_ws; (void)ws_size;
  biquad_eq_tdm_kernel<<<dim3(NBLOCKS), dim3(THREADS), 0, stream>>>(x, y);
}